// Unfold_72035191488644
// MI455X (gfx1250) — compile-verified
//
#include <hip/hip_runtime.h>
#include <stdint.h>

// Unfold (im2col, K=3, pad=1) for x: (16, 64, 128, 128) f32 -> (16, 576, 16384) f32.
// Pure data movement: stage each image tile into LDS once via the CDNA5 async
// global->LDS path, then fan out the 9 shifted copies with coalesced,
// non-temporal float4 stores. HBM traffic ~= 64 MB read + 604 MB write
// -> ~29 us roofline at 23.3 TB/s (write-dominated).

#define H_DIM   128
#define W_DIM   128
#define TILE_H  32
#define NROWS   34        // TILE_H + 2 halo rows
#define RSTRIDE 136       // floats per LDS row (544 B; data at [4..131] is 16B-aligned)
#define N_IMG   1024      // b*c = 16*64

typedef float v4f __attribute__((ext_vector_type(4)));

__global__ __launch_bounds__(256)
void unfold3x3_kernel(const float* __restrict__ x, float* __restrict__ out) {
    __shared__ __align__(16) float tile[NROWS * RSTRIDE];

    const int tid  = threadIdx.x;
    const int lane = tid & 31;
    const int wave = tid >> 5;

    const int img   = blockIdx.x >> 2;   // which (b,c) image
    const int tileY = blockIdx.x & 3;    // which 32-row band
    const int y0    = tileY * TILE_H;

    const float*   xg       = x + (size_t)img * (H_DIM * W_DIM);
    const uint32_t lds_base = (uint32_t)(uintptr_t)(&tile[0]);

    // Guard columns: xp col -1 (float idx 3) and xp col 128 (float idx 132)
    // of every LDS row are permanently zero (image-width padding).
    if (tid < 2 * NROWS) {
        int r = tid >> 1;
        int c = (tid & 1) ? 132 : 3;
        tile[r * RSTRIDE + c] = 0.0f;
    }

    // Stage halo tile rows. In-range rows: async DMA global->LDS, 512 B/row
    // (32 lanes x b128). Out-of-range rows (image top/bottom): zero via DS.
    for (int r = wave; r < NROWS; r += 8) {
        int gy = y0 - 1 + r;
        if (0 <= gy && gy < H_DIM) {
            uint32_t voff = (uint32_t)(gy * (W_DIM * 4) + lane * 16);
            uint32_t ldsa = lds_base + (uint32_t)(r * (RSTRIDE * 4) + 16 + lane * 16);
            asm volatile("global_load_async_to_lds_b128 %0, %1, %2"
                         :
                         : "v"(ldsa), "v"(voff), "s"(xg)
                         : "memory");
        } else {
            v4f z = (v4f)(0.0f);
            *(v4f*)&tile[r * RSTRIDE + lane * 4] = z;                 // floats 0..127
            if (lane < 2)
                *(v4f*)&tile[r * RSTRIDE + 128 + lane * 4] = z;       // floats 128..135
        }
    }
    asm volatile("s_wait_asynccnt 0x0" ::: "memory");  // my async writes landed in LDS
    __syncthreads();                                    // everyone's landed

    // Fan-out phase: 96 (row, vertical-shift) tasks per block, 12 per wave.
    // Each lane loads 6 consecutive floats and emits 3 overlapping float4s
    // (the three horizontal shifts), stored non-temporally to 3 out channels.
    for (int t = wave; t < TILE_H * 3; t += 8) {
        int yl = t / 3;                 // local output row, 0..31
        int i  = t - yl * 3;            // vertical kernel offset, 0..2
        int rb = (yl + i) * RSTRIDE + 3 + lane * 4;  // xp[y0+yl-1+i][4*lane-1]
        float f0 = tile[rb + 0], f1 = tile[rb + 1], f2 = tile[rb + 2];
        float f3 = tile[rb + 3], f4 = tile[rb + 4], f5 = tile[rb + 5];

        size_t ob = (size_t)(img * 9 + i * 3) * (size_t)(H_DIM * W_DIM)
                  + (size_t)(y0 + yl) * W_DIM + (size_t)(lane * 4);

        v4f o0 = {f0, f1, f2, f3};   // j = 0
        v4f o1 = {f1, f2, f3, f4};   // j = 1
        v4f o2 = {f2, f3, f4, f5};   // j = 2
        __builtin_nontemporal_store(o0, (v4f*)(out + ob));
        __builtin_nontemporal_store(o1, (v4f*)(out + ob + H_DIM * W_DIM));
        __builtin_nontemporal_store(o2, (v4f*)(out + ob + 2 * H_DIM * W_DIM));
    }
}

extern "C" void kernel_launch(void* const* d_in, const int* in_sizes, int n_in,
                              void* d_out, int out_size, void* d_ws, size_t ws_size,
                              hipStream_t stream) {
    (void)in_sizes; (void)n_in; (void)d_ws; (void)ws_size; (void)out_size;
    const float* x   = (const float*)d_in[0];
    float*       out = (float*)d_out;
    // 1024 images x 4 row-bands = 4096 blocks, 256 threads (8 wave32s) each.
    unfold3x3_kernel<<<dim3(N_IMG * 4), dim3(256), 0, stream>>>(x, out);
}